// GATLayer_4612794875975
// MI455X (gfx1250) — compile-verified
//
#include <hip/hip_runtime.h>
#include <math.h>

#define IN_DIM   128
#define OUT_DIM  64
#define REL_DIM  32
#define NEG_SLOPE 0.01f

typedef __attribute__((ext_vector_type(2))) float v2f;
typedef __attribute__((ext_vector_type(8))) float v8f;

#if __has_builtin(__builtin_amdgcn_wmma_f32_16x16x4_f32)
#define HAVE_WMMA_F32X4 1
#endif

// ---------------------------------------------------------------------------
// Fold the tiny projections into constants:
//   c3[0..31] = W_r @ a3,  c4 = W_s . a4,  c5 = W_t . a5
// cvec layout: [0..31]=c3, [32]=c4, [33]=c5
// ---------------------------------------------------------------------------
__global__ void k_const(const float* __restrict__ Wr, const float* __restrict__ Ws,
                        const float* __restrict__ Wt, const float* __restrict__ a,
                        float* __restrict__ cvec) {
    int t = threadIdx.x;
    if (t < REL_DIM) {
        const float* a3 = a + 2 * OUT_DIM;
        float s = 0.f;
        #pragma unroll
        for (int j = 0; j < OUT_DIM; ++j) s += Wr[t * OUT_DIM + j] * a3[j];
        cvec[t] = s;
    } else if (t == REL_DIM) {
        const float* a4 = a + 3 * OUT_DIM;
        float s = 0.f;
        #pragma unroll
        for (int j = 0; j < OUT_DIM; ++j) s += Ws[j] * a4[j];
        cvec[REL_DIM] = s;
    } else if (t == REL_DIM + 1) {
        const float* a5 = a + 4 * OUT_DIM;
        float s = 0.f;
        #pragma unroll
        for (int j = 0; j < OUT_DIM; ++j) s += Wt[j] * a5[j];
        cvec[REL_DIM + 1] = s;
    }
}

// ---------------------------------------------------------------------------
// z = h @ W_n via V_WMMA_F32_16X16X4_F32 (exact f32 FMA).
// One wave handles a 16-row tile, all 64 output columns (4 n-tiles).
// A 16x4 tile:  lane -> (M = lane&15, K = (lane>=16?2:0) + {vgpr0,vgpr1})
// B 4x16 tile:  lane -> (K = (lane>=16?2:0) + {vgpr0,vgpr1}, N = lane&15)
// C 16x16:      vgpr r, lane -> (M = r + (lane>=16?8:0), N = lane&15)
// ---------------------------------------------------------------------------
#ifdef HAVE_WMMA_F32X4
__global__ void k_node_gemm(const float* __restrict__ h, const float* __restrict__ Wn,
                            float* __restrict__ z, int n_nodes) {
    const int gwave = blockIdx.x * (blockDim.x >> 5) + (threadIdx.x >> 5);
    const int lane  = threadIdx.x & 31;
    const int row0  = gwave * 16;
    if (row0 >= n_nodes) return;   // wave-uniform: EXEC stays all-ones for WMMA

    const int m  = lane & 15;
    const int hi = lane >> 4;       // 0 or 1
    const int kh = hi << 1;         // 0 or 2

    v8f acc0 = {}, acc1 = {}, acc2 = {}, acc3 = {};

    const float* hrow = h + (size_t)(row0 + m) * IN_DIM;

    #pragma unroll 4
    for (int k0 = 0; k0 < IN_DIM; k0 += 4) {
        // A tile: two consecutive floats per lane -> single b64 load
        const float* ap = hrow + k0 + kh;
        v2f A; A.x = ap[0]; A.y = ap[1];

        // B tiles: rows k0+kh and k0+kh+1 of W_n (row-major 128x64)
        const float* b0 = Wn + (size_t)(k0 + kh) * OUT_DIM;
        const float* b1 = b0 + OUT_DIM;
        v2f B0; B0.x = b0[m];      B0.y = b1[m];
        v2f B1; B1.x = b0[m + 16]; B1.y = b1[m + 16];
        v2f B2; B2.x = b0[m + 32]; B2.y = b1[m + 32];
        v2f B3; B3.x = b0[m + 48]; B3.y = b1[m + 48];

        acc0 = __builtin_amdgcn_wmma_f32_16x16x4_f32(false, A, false, B0, (short)0, acc0, false, false);
        acc1 = __builtin_amdgcn_wmma_f32_16x16x4_f32(false, A, false, B1, (short)0, acc1, false, false);
        acc2 = __builtin_amdgcn_wmma_f32_16x16x4_f32(false, A, false, B2, (short)0, acc2, false, false);
        acc3 = __builtin_amdgcn_wmma_f32_16x16x4_f32(false, A, false, B3, (short)0, acc3, false, false);
    }

    float* zp = z + (size_t)row0 * OUT_DIM;
    #pragma unroll
    for (int r = 0; r < 8; ++r) {
        const int row = r + hi * 8;
        zp[(size_t)row * OUT_DIM +  0 + m] = acc0[r];
        zp[(size_t)row * OUT_DIM + 16 + m] = acc1[r];
        zp[(size_t)row * OUT_DIM + 32 + m] = acc2[r];
        zp[(size_t)row * OUT_DIM + 48 + m] = acc3[r];
    }
}
#else
__global__ void k_node_gemm_fallback(const float* __restrict__ h, const float* __restrict__ Wn,
                                     float* __restrict__ z, int n_nodes) {
    int i = blockIdx.x * blockDim.x + threadIdx.x;
    int node = i >> 6, col = i & 63;
    if (node >= n_nodes) return;
    float acc = 0.f;
    for (int k = 0; k < IN_DIM; ++k) acc += h[(size_t)node * IN_DIM + k] * Wn[k * OUT_DIM + col];
    z[(size_t)node * OUT_DIM + col] = acc;
}
#endif

// ---------------------------------------------------------------------------
// Per-node attention scalars: s1 = z . a1, s2 = z . a2
// ---------------------------------------------------------------------------
__global__ void k_s1s2(const float* __restrict__ z, const float* __restrict__ a,
                       float* __restrict__ s1, float* __restrict__ s2, int n_nodes) {
    int i = blockIdx.x * blockDim.x + threadIdx.x;
    if (i >= n_nodes) return;
    const float4* zr = (const float4*)(z + (size_t)i * OUT_DIM);
    const float4* a1 = (const float4*)(a);
    const float4* a2 = (const float4*)(a + OUT_DIM);
    float r1 = 0.f, r2 = 0.f;
    #pragma unroll
    for (int j = 0; j < OUT_DIM / 4; ++j) {
        float4 zv = zr[j], v1 = a1[j], v2 = a2[j];
        r1 += zv.x * v1.x + zv.y * v1.y + zv.z * v1.z + zv.w * v1.w;
        r2 += zv.x * v2.x + zv.y * v2.y + zv.z * v2.z + zv.w * v2.w;
    }
    s1[i] = r1;
    s2[i] = r2;
}

// ---------------------------------------------------------------------------
// Init: segmax = -inf, segsum = 0, out = 0
// ---------------------------------------------------------------------------
__global__ void k_init(float* __restrict__ segmax, float* __restrict__ segsum,
                       float* __restrict__ out, int n_nodes) {
    int i = blockIdx.x * blockDim.x + threadIdx.x;
    if (i < n_nodes * OUT_DIM) out[i] = 0.f;
    if (i < n_nodes) { segmax[i] = -__builtin_inff(); segsum[i] = 0.f; }
}

// ---------------------------------------------------------------------------
// Pass 1: edge logit + leaky-relu + atomic segment max (sign-aware punning)
// ---------------------------------------------------------------------------
__device__ __forceinline__ void atomic_max_float(float* addr, float val) {
    if (val >= 0.f) atomicMax((int*)addr, __float_as_int(val));
    else            atomicMin((unsigned int*)addr, __float_as_uint(val));
}

__global__ void k_logit(const float* __restrict__ rel, const float* __restrict__ score,
                        const float* __restrict__ ts,  const int* __restrict__ src,
                        const int* __restrict__ dst,   const float* __restrict__ s1,
                        const float* __restrict__ s2,  const float* __restrict__ cvec,
                        float* __restrict__ logit, float* __restrict__ segmax, int n_edges) {
    int e = blockIdx.x * blockDim.x + threadIdx.x;
    if (e >= n_edges) return;
    const float4* rp = (const float4*)(rel + (size_t)e * REL_DIM);
    const float4* cp = (const float4*)cvec;
    float acc = 0.f;
    #pragma unroll
    for (int j = 0; j < REL_DIM / 4; ++j) {
        float4 rv = rp[j], cv = cp[j];
        acc += rv.x * cv.x + rv.y * cv.y + rv.z * cv.z + rv.w * cv.w;
    }
    int d = dst[e];
    float l = s1[src[e]] + s2[d] + acc + score[e] * cvec[REL_DIM] + ts[e] * cvec[REL_DIM + 1];
    l = (l >= 0.f) ? l : NEG_SLOPE * l;
    logit[e] = l;
    atomic_max_float(segmax + d, l);
}

// ---------------------------------------------------------------------------
// Pass 2: ex = exp(logit - segmax[dst]); segsum[dst] += ex
// ---------------------------------------------------------------------------
__global__ void k_expsum(const float* __restrict__ logit, const int* __restrict__ dst,
                         const float* __restrict__ segmax, float* __restrict__ ex,
                         float* __restrict__ segsum, int n_edges) {
    int e = blockIdx.x * blockDim.x + threadIdx.x;
    if (e >= n_edges) return;
    int d = dst[e];
    float v = expf(logit[e] - segmax[d]);
    ex[e] = v;
    atomicAdd(segsum + d, v);
}

// ---------------------------------------------------------------------------
// Pass 3: out[dst] += (ex/segsum[dst]) * z[src]
// 64 consecutive threads per edge -> coalesced row reads and row atomics
// ---------------------------------------------------------------------------
__global__ void k_scatter(const float* __restrict__ z, const int* __restrict__ src,
                          const int* __restrict__ dst, const float* __restrict__ ex,
                          const float* __restrict__ segsum, float* __restrict__ out,
                          int n_edges) {
    long long gid = (long long)blockIdx.x * blockDim.x + threadIdx.x;
    int e = (int)(gid >> 6);
    int j = (int)(gid & 63);
    if (e >= n_edges) return;
    int d = dst[e];
    float alpha = ex[e] / segsum[d];
    atomicAdd(out + (size_t)d * OUT_DIM + j, alpha * z[(size_t)src[e] * OUT_DIM + j]);
}

// ---------------------------------------------------------------------------
extern "C" void kernel_launch(void* const* d_in, const int* in_sizes, int n_in,
                              void* d_out, int out_size, void* d_ws, size_t ws_size,
                              hipStream_t stream) {
    const float* h    = (const float*)d_in[0];
    const float* rel  = (const float*)d_in[1];
    const float* scr  = (const float*)d_in[2];
    const float* ts   = (const float*)d_in[3];
    const int*   src  = (const int*)d_in[4];
    const int*   dst  = (const int*)d_in[5];
    const float* Wn   = (const float*)d_in[6];
    const float* Wr   = (const float*)d_in[7];
    const float* Ws   = (const float*)d_in[8];
    const float* Wt   = (const float*)d_in[9];
    const float* a    = (const float*)d_in[10];
    float* out        = (float*)d_out;

    const int N = in_sizes[0] / IN_DIM;
    const int E = in_sizes[4];

    float* ws = (float*)d_ws;
    size_t off = 0;
    float* z      = ws + off; off += (size_t)N * OUT_DIM;
    float* s1     = ws + off; off += N;
    float* s2     = ws + off; off += N;
    float* segmax = ws + off; off += N;
    float* segsum = ws + off; off += N;
    float* logit  = ws + off; off += E;
    float* ex     = ws + off; off += E;
    float* cvec   = ws + off; off += 64;

    // constants (tiny)
    k_const<<<1, 64, 0, stream>>>(Wr, Ws, Wt, a, cvec);

    // init segment buffers + output
    {
        int total = N * OUT_DIM;
        k_init<<<(total + 255) / 256, 256, 0, stream>>>(segmax, segsum, out, N);
    }

    // node GEMM (WMMA f32 16x16x4)
#ifdef HAVE_WMMA_F32X4
    {
        int n_waves = (N + 15) / 16;          // one wave per 16-row tile
        int wpb = 4;                           // 128-thread blocks
        int blocks = (n_waves + wpb - 1) / wpb;
        k_node_gemm<<<blocks, wpb * 32, 0, stream>>>(h, Wn, z, N);
    }
#else
    {
        int total = N * OUT_DIM;
        k_node_gemm_fallback<<<(total + 255) / 256, 256, 0, stream>>>(h, Wn, z, N);
    }
#endif

    // per-node attention scalars
    k_s1s2<<<(N + 255) / 256, 256, 0, stream>>>(z, a, s1, s2, N);

    // pass 1: logits + segment max
    k_logit<<<(E + 255) / 256, 256, 0, stream>>>(rel, scr, ts, src, dst, s1, s2,
                                                 cvec, logit, segmax, E);

    // pass 2: exp + segment sum
    k_expsum<<<(E + 255) / 256, 256, 0, stream>>>(logit, dst, segmax, ex, segsum, E);

    // pass 3: weighted scatter into output
    {
        long long total = (long long)E * OUT_DIM;
        int blocks = (int)((total + 255) / 256);
        k_scatter<<<blocks, 256, 0, stream>>>(z, src, dst, ex, segsum, out, E);
    }
}